// HybridXGModel_72722386256530
// MI455X (gfx1250) — compile-verified
//
#include <hip/hip_runtime.h>
#include <hip/hip_bf16.h>

typedef __attribute__((ext_vector_type(2))) float v2f;
typedef __attribute__((ext_vector_type(8))) float v8f;

#define HID 128
#define META 27

// ---------------------------------------------------------------------------
// utility fills
// ---------------------------------------------------------------------------
__global__ void k_fill(float* __restrict__ p, float v, int n) {
    int i = blockIdx.x * blockDim.x + threadIdx.x;
    if (i < n) p[i] = v;
}

// deg[dst] += 1  (deg pre-filled with 1.0 for the self loop)
__global__ void k_degree(const long long* __restrict__ dst, float* __restrict__ deg, int e) {
    int i = blockIdx.x * blockDim.x + threadIdx.x;
    if (i < e) atomicAdd(&deg[dst[i]], 1.0f);
}

// in-place deg -> rsqrt(max(deg,1))
__global__ void k_rsqrt(float* __restrict__ d, int n) {
    int i = blockIdx.x * blockDim.x + threadIdx.x;
    if (i < n) d[i] = __frsqrt_rn(fmaxf(d[i], 1.0f));
}

// ---------------------------------------------------------------------------
// GEMM + scale epilogue:  G[i,:] = dinv[i] * (Hin @ W)[i,:];  acc = G (self loop)
// Hin: [n,128] row-major, W: [128,128] row-major.
// 256 threads = 8 waves; each wave computes a 16x128 output strip via
// v_wmma_f32_16x16x4_f32 (8 col tiles x 32 k-steps).
// ---------------------------------------------------------------------------
__global__ __launch_bounds__(256) void k_gemm_scale(
    const float* __restrict__ Hin, const float* __restrict__ W,
    const float* __restrict__ dinv, float* __restrict__ G,
    float* __restrict__ acc, int nrows)
{
    __shared__ float sW[HID * HID];          // 64 KB of the 320 KB WGP LDS
    for (int i = threadIdx.x; i < HID * HID; i += 256) sW[i] = W[i];
    __syncthreads();

    const int lane  = threadIdx.x & 31;
    const int wave  = threadIdx.x >> 5;              // 0..7
    const int row0  = blockIdx.x * 128 + wave * 16;  // strip start
    const int m     = lane & 15;                     // M index within tile
    const int khalf = lane >> 4;                     // 0 -> K=0,1 ; 1 -> K=2,3
    const int row   = row0 + m;
    const int crow  = row < nrows ? row : (nrows - 1);   // clamped: loads stay valid
    const float msk = row < nrows ? 1.0f : 0.0f;         // branchless, EXEC stays ~0
    const float* __restrict__ arow = Hin + (size_t)crow * HID;

    // Preload A fragments for all 32 k-steps (kk = 4s + 2*khalf, pairs).
    v2f afrag[32];
#pragma unroll
    for (int s = 0; s < 32; ++s) {
        const int kk = 4 * s + 2 * khalf;
        const float2 av = *(const float2*)(arow + kk);
        afrag[s].x = av.x * msk;
        afrag[s].y = av.y * msk;
    }

    for (int ct = 0; ct < 8; ++ct) {
        const int col = ct * 16 + m;
        v8f c = {};
#pragma unroll
        for (int s = 0; s < 32; ++s) {
            const int kk = 4 * s + 2 * khalf;
            v2f b;
            b.x = sW[kk * HID + col];
            b.y = sW[(kk + 1) * HID + col];
            // 8 args: (neg_a, A, neg_b, B, c_mod, C, reuse_a, reuse_b)
            c = __builtin_amdgcn_wmma_f32_16x16x4_f32(
                    false, afrag[s], false, b, (short)0, c, false, false);
        }
        // D layout: VGPR r -> M = r + 8*khalf (lanes 0-15: M=r, 16-31: M=8+r)
#pragma unroll
        for (int r = 0; r < 8; ++r) {
            const int orow = row0 + r + 8 * khalf;
            if (orow < nrows) {
                const float v = c[r] * dinv[orow];
                const size_t off = (size_t)orow * HID + col;
                G[off]   = v;
                acc[off] = v;      // self-loop contribution baked in
            }
        }
    }
}

// ---------------------------------------------------------------------------
// Edge scatter: acc[dst,:] += G[src,:]   (one 32-lane wave per edge, float4/lane)
// ---------------------------------------------------------------------------
__global__ void k_scatter(const float* __restrict__ G,
                          const long long* __restrict__ src,
                          const long long* __restrict__ dst,
                          float* __restrict__ acc, int nedges)
{
    int tid = blockIdx.x * blockDim.x + threadIdx.x;
    int e   = tid >> 5;
    if (e >= nedges) return;
    int c = (tid & 31) * 4;
    long long s = src[e];
    long long d = dst[e];
    const float4 g = *(const float4*)(G + (size_t)s * HID + c);
    float* a = acc + (size_t)d * HID + c;
    atomicAdd(a + 0, g.x);
    atomicAdd(a + 1, g.y);
    atomicAdd(a + 2, g.z);
    atomicAdd(a + 3, g.w);
}

// h[i,c] = relu(dinv[i]*acc[i,c] + bias[c])
__global__ void k_post(const float* __restrict__ acc, const float* __restrict__ dinv,
                       const float* __restrict__ bias, float* __restrict__ h, int n)
{
    int tid = blockIdx.x * blockDim.x + threadIdx.x;
    if (tid >= n * HID) return;
    int i = tid >> 7, c = tid & 127;
    float v = fmaf(dinv[i], acc[tid], bias[c]);
    h[tid] = v > 0.0f ? v : 0.0f;
}

// counts[batch[i]] += 1
__global__ void k_count(const long long* __restrict__ batch, float* __restrict__ counts, int n) {
    int i = blockIdx.x * blockDim.x + threadIdx.x;
    if (i < n) atomicAdd(&counts[batch[i]], 1.0f);
}

// summed[batch[i],c] += h[i,c]
__global__ void k_pool(const float* __restrict__ h, const long long* __restrict__ batch,
                       float* __restrict__ summed, int n)
{
    int tid = blockIdx.x * blockDim.x + threadIdx.x;
    if (tid >= n * HID) return;
    int i = tid >> 7, c = tid & 127;
    atomicAdd(&summed[(size_t)batch[i] * HID + c], h[tid]);
}

// ---------------------------------------------------------------------------
// MLP head: one 128-thread block per graph.
// z = [summed/count | meta] (155), hcol = relu(z @ Wh1 + bh1), out = hcol @ Wh2 + bh2
// ---------------------------------------------------------------------------
__global__ __launch_bounds__(128) void k_head(
    const float* __restrict__ summed, const float* __restrict__ counts,
    const float* __restrict__ meta,
    const float* __restrict__ Wh1, const float* __restrict__ bh1,
    const float* __restrict__ Wh2, const float* __restrict__ bh2,
    float* __restrict__ out)
{
    __shared__ float z[HID + META];
    __shared__ float red[HID];
    const int g = blockIdx.x, t = threadIdx.x;
    const float inv = 1.0f / fmaxf(counts[g], 1.0f);
    z[t] = summed[(size_t)g * HID + t] * inv;
    if (t < META) z[HID + t] = meta[g * META + t];
    __syncthreads();

    float s = bh1[t];
#pragma unroll 5
    for (int k = 0; k < HID + META; ++k)
        s = fmaf(z[k], Wh1[k * HID + t], s);
    s = s > 0.0f ? s : 0.0f;
    red[t] = s * Wh2[t];
    __syncthreads();
    for (int off = 64; off > 0; off >>= 1) {
        if (t < off) red[t] += red[t + off];
        __syncthreads();
    }
    if (t == 0) out[g] = red[0] + bh2[0];
}

// ---------------------------------------------------------------------------
extern "C" void kernel_launch(void* const* d_in, const int* in_sizes, int n_in,
                              void* d_out, int out_size, void* d_ws, size_t ws_size,
                              hipStream_t stream)
{
    const float*     x     = (const float*)d_in[0];
    const long long* ei    = (const long long*)d_in[1];   // [2,E] int64
    const long long* batch = (const long long*)d_in[2];   // [N] int64
    const float*     meta  = (const float*)d_in[3];
    const float*     Wl[3] = { (const float*)d_in[4], (const float*)d_in[6], (const float*)d_in[8] };
    const float*     bl[3] = { (const float*)d_in[5], (const float*)d_in[7], (const float*)d_in[9] };
    const float*     Wh1   = (const float*)d_in[10];
    const float*     bh1   = (const float*)d_in[11];
    const float*     Wh2   = (const float*)d_in[12];
    const float*     bh2   = (const float*)d_in[13];
    float*           out   = (float*)d_out;

    const int N = in_sizes[0] / HID;
    const int E = in_sizes[1] / 2;
    const int B = in_sizes[3] / META;

    const long long* src = ei;
    const long long* dst = ei + E;

    // workspace carve-up (float units; keep 16B alignment)
    float* ws     = (float*)d_ws;
    size_t nAlign = (size_t)((N + 3) & ~3);
    float* dinv   = ws;                              // N  (doubles as deg)
    float* hbuf   = dinv + nAlign;                   // N*128
    float* G      = hbuf + (size_t)N * HID;          // N*128
    float* acc    = G    + (size_t)N * HID;          // N*128
    float* summed = acc  + (size_t)N * HID;          // B*128
    float* counts = summed + (size_t)B * HID;        // B

    const int T = 256;
    // degree & normalization
    k_fill  <<<(N + T - 1) / T, T, 0, stream>>>(dinv, 1.0f, N);
    k_degree<<<(E + T - 1) / T, T, 0, stream>>>(dst, dinv, E);
    k_rsqrt <<<(N + T - 1) / T, T, 0, stream>>>(dinv, N);

    // 3 GCN layers
    const int gemmGrid = (N + 127) / 128;
    const int scatGrid = (int)(((size_t)E * 32 + T - 1) / T);
    const int nodeGrid = (int)(((size_t)N * HID + T - 1) / T);
    const float* hin = x;
    for (int l = 0; l < 3; ++l) {
        k_gemm_scale<<<gemmGrid, 256, 0, stream>>>(hin, Wl[l], dinv, G, acc, N);
        k_scatter   <<<scatGrid, T, 0, stream>>>(G, src, dst, acc, E);
        k_post      <<<nodeGrid, T, 0, stream>>>(acc, dinv, bl[l], hbuf, N);
        hin = hbuf;
    }

    // global mean pool
    k_fill <<<(B * HID + T - 1) / T, T, 0, stream>>>(summed, 0.0f, B * HID);
    k_fill <<<(B + T - 1) / T, T, 0, stream>>>(counts, 0.0f, B);
    k_count<<<(N + T - 1) / T, T, 0, stream>>>(batch, counts, N);
    k_pool <<<nodeGrid, T, 0, stream>>>(hbuf, batch, summed, N);

    // head
    k_head<<<B, 128, 0, stream>>>(summed, counts, meta, Wh1, bh1, Wh2, bh2, out);
}